// GCN_37203006718671
// MI455X (gfx1250) — compile-verified
//
#include <hip/hip_runtime.h>
#include <math.h>

typedef __attribute__((ext_vector_type(2))) float v2f;
typedef __attribute__((ext_vector_type(8))) float v8f;

#define NN   4096
#define BB   32
#define HH   256
#define TD   128

// ---------------------------------------------------------------------------
// Coefficients c_m = sum_{j+k=m} sum_i w_g1[0,i,j]*w_g2[i,0,k], m=0..4
// ---------------------------------------------------------------------------
__global__ void k_coef(const float* __restrict__ wg1, const float* __restrict__ wg2,
                       float* __restrict__ c) {
  __shared__ float acc9[9];
  const int tid = threadIdx.x;           // 256 threads, one per channel i
  if (tid < 9) acc9[tid] = 0.f;
  __syncthreads();
  const float a0 = wg1[tid * 3 + 0], a1 = wg1[tid * 3 + 1], a2 = wg1[tid * 3 + 2];
  const float b0 = wg2[tid * 3 + 0], b1 = wg2[tid * 3 + 1], b2 = wg2[tid * 3 + 2];
  atomicAdd(&acc9[0], a0 * b0); atomicAdd(&acc9[1], a0 * b1); atomicAdd(&acc9[2], a0 * b2);
  atomicAdd(&acc9[3], a1 * b0); atomicAdd(&acc9[4], a1 * b1); atomicAdd(&acc9[5], a1 * b2);
  atomicAdd(&acc9[6], a2 * b0); atomicAdd(&acc9[7], a2 * b1); atomicAdd(&acc9[8], a2 * b2);
  __syncthreads();
  if (tid == 0) {
    float cc[5] = {0.f, 0.f, 0.f, 0.f, 0.f};
    for (int j = 0; j < 3; ++j)
      for (int k = 0; k < 3; ++k)
        cc[j + k] += acc9[j * 3 + k];
    for (int m = 0; m < 5; ++m) c[m] = cc[m];
  }
}

// ---------------------------------------------------------------------------
// X0[n][b] = x[b][n]   (column-major batch layout for the hop GEMMs)
// ---------------------------------------------------------------------------
__global__ void k_transpose(const float* __restrict__ x, float* __restrict__ X0) {
  const int i = blockIdx.x * blockDim.x + threadIdx.x;   // 131072 total
  const int b = i & 31, n = i >> 5;
  X0[n * BB + b] = x[b * NN + n];
}

// ---------------------------------------------------------------------------
// Hop: Y[4096,32] += L[4096,4096] @ X[4096,32]  (fp32 WMMA 16x16x4)
// grid = (64 M-strips, 4 K-splits), block = 256 (8 waves)
// wave w: m-subtile = w>>1 (0..3), n-subtile = w&1 (0..1)
// Double-buffered LDS fed by GLOBAL_LOAD_ASYNC_TO_LDS_B128 (ASYNCcnt),
// overlapping the next strip's DMA with the current strip's WMMAs.
// ---------------------------------------------------------------------------
#define A_S 68   // padded LDS row stride (floats) for 64x64 A strip
#define B_S 40   // padded LDS row stride (floats) for 64x32 B strip

__global__ __launch_bounds__(256) void k_hop(const float* __restrict__ L,
                                             const float* __restrict__ Xin,
                                             float* __restrict__ Y) {
  __shared__ __align__(16) float As[2][64 * A_S];
  __shared__ __align__(16) float Bs[2][64 * B_S];
  const int tid  = threadIdx.x;
  const int lane = tid & 31;
  const int wave = tid >> 5;
  const int mt   = wave >> 1;
  const int nt   = wave & 1;
  const int m0   = blockIdx.x * 64;
  const int k0b  = blockIdx.y * 1024;
  const int lm   = lane & 15;
  const int klo  = (lane >> 4) << 1;   // 0 or 2

  // issue one 64x64 A strip + 64x32 B strip as async global->LDS DMA
  // (6 x B128 per thread => wave ASYNCcnt advances by 6 per strip)
  auto issue_strip = [&](int k0, int buf) {
#pragma unroll
    for (int it = 0; it < 4; ++it) {
      const int f   = it * 256 + tid;
      const int row = f >> 4;
      const int c4  = (f & 15) << 2;
      const float* g = &L[(size_t)(m0 + row) * NN + k0 + c4];
      // low 32 bits of a generic __shared__ pointer == LDS byte address
      const unsigned lds = (unsigned)(uintptr_t)&As[buf][row * A_S + c4];
      asm volatile("global_load_async_to_lds_b128 %0, %1, off"
                   :: "v"(lds), "v"(g) : "memory");
    }
#pragma unroll
    for (int it = 0; it < 2; ++it) {
      const int f   = it * 256 + tid;
      const int row = f >> 3;
      const int c4  = (f & 7) << 2;
      const float* g = &Xin[(size_t)(k0 + row) * BB + c4];
      const unsigned lds = (unsigned)(uintptr_t)&Bs[buf][row * B_S + c4];
      asm volatile("global_load_async_to_lds_b128 %0, %1, off"
                   :: "v"(lds), "v"(g) : "memory");
    }
  };

  v8f acc = {};
  issue_strip(k0b, 0);                       // prologue: strip 0 -> buffer 0
  for (int s = 0; s < 16; ++s) {
    const int cur = s & 1;
    if (s + 1 < 16) {
      issue_strip(k0b + (s + 1) * 64, cur ^ 1);
      // retire the previous 6 async ops (current strip); next 6 stay in flight
      asm volatile("s_wait_asynccnt 0x6" ::: "memory");
    } else {
      asm volatile("s_wait_asynccnt 0x0" ::: "memory");
    }
    __syncthreads();                         // all waves' strip data in LDS

    const float* Ab = &As[cur][0];
    const float* Bb = &Bs[cur][0];
#pragma unroll
    for (int kk = 0; kk < 64; kk += 4) {
      v2f a, b;
      const float* ap = &Ab[(mt * 16 + lm) * A_S + kk + klo];
      a.x = ap[0];
      a.y = ap[1];
      b.x = Bb[(kk + klo) * B_S + nt * 16 + lm];
      b.y = Bb[(kk + klo + 1) * B_S + nt * 16 + lm];
      acc = __builtin_amdgcn_wmma_f32_16x16x4_f32(false, a, false, b, (short)0, acc,
                                                  false, false);
    }
    __syncthreads();                         // done reading before DMA overwrites
  }
  const int col   = nt * 16 + lm;
  const int rbase = m0 + mt * 16 + ((lane >> 4) << 3);
#pragma unroll
  for (int i = 0; i < 8; ++i)
    atomicAdd(&Y[(size_t)(rbase + i) * BB + col], acc[i]);
}

// ---------------------------------------------------------------------------
// G[i] = c0*X0 + c1*Y1 + c2*Y2 + c3*Y3 + c4*Y4
// ---------------------------------------------------------------------------
__global__ void k_combine(const float* __restrict__ X0, const float* __restrict__ Y1,
                          const float* __restrict__ Y2, const float* __restrict__ Y3,
                          const float* __restrict__ Y4, const float* __restrict__ c,
                          float* __restrict__ G, int n) {
  const int i = blockIdx.x * blockDim.x + threadIdx.x;
  if (i < n)
    G[i] = c[0] * X0[i] + c[1] * Y1[i] + c[2] * Y2[i] + c[3] * Y3[i] + c[4] * Y4[i];
}

// ---------------------------------------------------------------------------
// Generic small GEMM, M = 32 fixed:  C[32,N] += A @ B[K,N]  (fp32 WMMA)
// a_trans=0: A row-major [32, a_stride]; a_trans=1: A[m][k] = A[k*a_stride + m]
// grid = (N/16, ksplit), block = 64 (wave 0 -> rows 0-15, wave 1 -> rows 16-31)
// ---------------------------------------------------------------------------
__global__ __launch_bounds__(64) void k_gemm32(const float* __restrict__ A, int a_stride,
                                               int a_trans, const float* __restrict__ B,
                                               int N, float* __restrict__ C, int K) {
  const int lane = threadIdx.x & 31;
  const int m0   = (threadIdx.x >> 5) * 16;
  const int n0   = blockIdx.x * 16;
  const int klen = K / gridDim.y;
  const int k0   = blockIdx.y * klen;
  const int lm   = lane & 15;
  const int klo  = (lane >> 4) << 1;
  const int m    = m0 + lm;
  const int n    = n0 + lm;

  v8f acc = {};
  for (int k = k0; k < k0 + klen; k += 4) {
    const int kl = k + klo;
    v2f a, b;
    if (a_trans) {
      a.x = A[(size_t)kl * a_stride + m];
      a.y = A[(size_t)(kl + 1) * a_stride + m];
    } else {
      a.x = A[(size_t)m * a_stride + kl];
      a.y = A[(size_t)m * a_stride + kl + 1];
    }
    b.x = B[(size_t)kl * N + n];
    b.y = B[(size_t)(kl + 1) * N + n];
    acc = __builtin_amdgcn_wmma_f32_16x16x4_f32(false, a, false, b, (short)0, acc,
                                                false, false);
  }
  const int rbase = m0 + ((lane >> 4) << 3);
#pragma unroll
  for (int i = 0; i < 8; ++i)
    atomicAdd(&C[(size_t)(rbase + i) * N + n], acc[i]);
}

// ---------------------------------------------------------------------------
// Elementwise epilogues
// ---------------------------------------------------------------------------
__global__ void k_bias(float* __restrict__ C, const float* __restrict__ bias, int mask,
                       int total, int act) {
  const int i = blockIdx.x * blockDim.x + threadIdx.x;
  if (i >= total) return;
  float v = C[i] + bias[i & mask];
  if (act) v = v / (1.f + expf(-v));   // SiLU
  C[i] = v;
}

__global__ void k_resmix(float* __restrict__ h, const float* __restrict__ z, int total) {
  const int i = blockIdx.x * blockDim.x + threadIdx.x;
  if (i < total) h[i] = (h[i] + z[i]) * 0.70710678118654752f;
}

__global__ void k_add(const float* __restrict__ a, const float* __restrict__ b,
                      float* __restrict__ o, int total) {
  const int i = blockIdx.x * blockDim.x + threadIdx.x;
  if (i < total) o[i] = a[i] + b[i];
}

// temb[b][d]: d<64 -> cos(t[b]*f_d), d>=64 -> sin(t[b]*f_{d-64})
__global__ void k_temb(const float* __restrict__ t, float* __restrict__ e) {
  const int i = blockIdx.x * blockDim.x + threadIdx.x;   // 32*128
  const int b = i >> 7, d = i & 127;
  const int hd = d & 63;
  const float freq = expf(-logf(10000.f) * (float)hd / 64.f);
  const float arg  = t[b] * freq;
  e[i] = (d < 64) ? cosf(arg) : sinf(arg);
}

// ---------------------------------------------------------------------------
extern "C" void kernel_launch(void* const* d_in, const int* in_sizes, int n_in,
                              void* d_out, int out_size, void* d_ws, size_t ws_size,
                              hipStream_t stream) {
  (void)in_sizes; (void)n_in; (void)out_size; (void)ws_size;
  const float* x    = (const float*)d_in[0];
  const float* t    = (const float*)d_in[1];
  const float* Lap  = (const float*)d_in[2];
  const float* wg1  = (const float*)d_in[3];
  const float* wg2  = (const float*)d_in[4];
  const float* wt1  = (const float*)d_in[5];
  const float* bt1  = (const float*)d_in[6];
  const float* wt2  = (const float*)d_in[7];
  const float* bt2  = (const float*)d_in[8];
  const float* wmap = (const float*)d_in[9];
  const float* bmap = (const float*)d_in[10];
  const float* resw = (const float*)d_in[11];
  const float* resb = (const float*)d_in[12];
  const float* wo1  = (const float*)d_in[13];
  const float* bo1  = (const float*)d_in[14];
  const float* wo2  = (const float*)d_in[15];
  const float* bo2  = (const float*)d_in[16];
  float* out = (float*)d_out;

  const int NB = NN * BB;          // 131072
  float* ws   = (float*)d_ws;
  float* X0   = ws;                float* Y1 = X0 + NB;
  float* Y2   = Y1 + NB;           float* Y3 = Y2 + NB;
  float* Y4   = Y3 + NB;           float* G  = Y4 + NB;
  float* c5   = G + NB;            // 8 floats
  float* temb = c5 + 8;            // 32*128
  float* T1   = temb + BB * TD;    // 32*256
  float* tout = T1 + BB * HH;
  float* h    = tout + BB * HH;
  float* zA   = h + BB * HH;
  float* zB   = zA + BB * HH;
  float* u    = zB + BB * HH;
  float* v    = u + BB * HH;

  const size_t nbBytes = (size_t)NB * sizeof(float);
  const size_t hBytes  = (size_t)BB * HH * sizeof(float);

  // ---- GCN block: out = sum_m c_m * L^m x ------------------------------
  k_coef<<<1, 256, 0, stream>>>(wg1, wg2, c5);
  k_transpose<<<NB / 256, 256, 0, stream>>>(x, X0);

  hipMemsetAsync(Y1, 0, nbBytes, stream);
  k_hop<<<dim3(64, 4), 256, 0, stream>>>(Lap, X0, Y1);
  hipMemsetAsync(Y2, 0, nbBytes, stream);
  k_hop<<<dim3(64, 4), 256, 0, stream>>>(Lap, Y1, Y2);
  hipMemsetAsync(Y3, 0, nbBytes, stream);
  k_hop<<<dim3(64, 4), 256, 0, stream>>>(Lap, Y2, Y3);
  hipMemsetAsync(Y4, 0, nbBytes, stream);
  k_hop<<<dim3(64, 4), 256, 0, stream>>>(Lap, Y3, Y4);
  k_combine<<<NB / 256, 256, 0, stream>>>(X0, Y1, Y2, Y3, Y4, c5, G, NB);

  // ---- h = G' @ w_map + b_map  (K=4096, 8-way K-split) -----------------
  hipMemsetAsync(h, 0, hBytes, stream);
  k_gemm32<<<dim3(HH / 16, 8), 64, 0, stream>>>(G, BB, 1, wmap, HH, h, NN);
  k_bias<<<32, 256, 0, stream>>>(h, bmap, HH - 1, BB * HH, 0);

  // ---- time embedding path --------------------------------------------
  k_temb<<<BB * TD / 256, 256, 0, stream>>>(t, temb);
  hipMemsetAsync(T1, 0, hBytes, stream);
  k_gemm32<<<dim3(HH / 16, 1), 64, 0, stream>>>(temb, TD, 0, wt1, HH, T1, TD);
  k_bias<<<32, 256, 0, stream>>>(T1, bt1, HH - 1, BB * HH, 1);
  hipMemsetAsync(tout, 0, hBytes, stream);
  k_gemm32<<<dim3(HH / 16, 1), 64, 0, stream>>>(T1, HH, 0, wt2, HH, tout, HH);
  k_bias<<<32, 256, 0, stream>>>(tout, bt2, HH - 1, BB * HH, 0);

  // ---- ResNet_FC -------------------------------------------------------
  float* za = zA;
  float* zb = zB;
  for (int r = 0; r < 2; ++r) {
    hipMemcpyAsync(za, h, hBytes, hipMemcpyDeviceToDevice, stream);
    for (int i = 0; i < 3; ++i) {
      hipMemsetAsync(zb, 0, hBytes, stream);
      k_gemm32<<<dim3(HH / 16, 1), 64, 0, stream>>>(
          za, HH, 0, resw + (size_t)(r * 3 + i) * HH * HH, HH, zb, HH);
      k_bias<<<32, 256, 0, stream>>>(zb, resb + (size_t)(r * 3 + i) * HH, HH - 1,
                                     BB * HH, 1);
      float* tmp = za; za = zb; zb = tmp;
    }
    k_resmix<<<32, 256, 0, stream>>>(h, za, BB * HH);
  }

  // ---- output head -----------------------------------------------------
  k_add<<<32, 256, 0, stream>>>(h, tout, u, BB * HH);
  hipMemsetAsync(v, 0, hBytes, stream);
  k_gemm32<<<dim3(HH / 16, 1), 64, 0, stream>>>(u, HH, 0, wo1, HH, v, HH);
  k_bias<<<32, 256, 0, stream>>>(v, bo1, HH - 1, BB * HH, 1);
  hipMemsetAsync(out, 0, nbBytes, stream);
  k_gemm32<<<dim3(NN / 16, 1), 64, 0, stream>>>(v, HH, 0, wo2, NN, out, HH);
  k_bias<<<NB / 256, 256, 0, stream>>>(out, bo2, NN - 1, NB, 0);
}